// MMD_Loss_43782896615716
// MI455X (gfx1250) — compile-verified
//
#include <hip/hip_runtime.h>

typedef __attribute__((ext_vector_type(2))) float v2f;
typedef __attribute__((ext_vector_type(8))) float v8f;

#define N_SRC 4096
#define N_TOT 8192
#define DIMS  256
#define EPS   1e-6f

__device__ __forceinline__ const float* row_ptr(const float* s, const float* t, int row) {
    return (row < N_SRC) ? (s + (size_t)row * DIMS)
                         : (t + (size_t)(row - N_SRC) * DIMS);
}

// ---------------------------------------------------------------------------
// Kernel 1: per-column partial sums. 64 blocks x 256 threads; block b owns
// 128 rows, thread t owns column t. Fully writes partial[64*256].
// ---------------------------------------------------------------------------
__global__ void mmd_colsum_kernel(const float* __restrict__ s,
                                  const float* __restrict__ t,
                                  float* __restrict__ partial) {
    const int tid = threadIdx.x;
    const int b   = blockIdx.x;
    const int r0  = b * (N_TOT / 64);   // 128 rows per block
    float acc = 0.f;
    for (int r = 0; r < N_TOT / 64; ++r) {
        acc += row_ptr(s, t, r0 + r)[tid];
    }
    partial[b * DIMS + tid] = acc;
}

// ---------------------------------------------------------------------------
// Kernel 2: row squared norms. One wave32 per row; 8 rows per 256-thr block.
// ---------------------------------------------------------------------------
__global__ void mmd_rowsq_kernel(const float* __restrict__ s,
                                 const float* __restrict__ t,
                                 float* __restrict__ sq) {
    const int wave = threadIdx.x >> 5;
    const int lane = threadIdx.x & 31;
    const int row  = blockIdx.x * 8 + wave;
    const float* p = row_ptr(s, t, row);
    float acc = 0.f;
    #pragma unroll
    for (int k = lane; k < DIMS; k += 32) {
        float x = p[k];
        acc += x * x;
    }
    #pragma unroll
    for (int off = 16; off > 0; off >>= 1)
        acc += __shfl_xor(acc, off, 32);
    if (lane == 0) sq[row] = acc;
}

// ---------------------------------------------------------------------------
// Kernel 3: bandwidth = (2n*sum(sq) - 2*||colsum||^2) / (n^2-n) / 4
// (KERNEL_MUL^(KERNEL_NUM//2) = 2^2 = 4).  Single block.
// ---------------------------------------------------------------------------
__global__ void mmd_bw_kernel(const float* __restrict__ partial,
                              const float* __restrict__ sq,
                              float* __restrict__ bw_out) {
    __shared__ float red[256];
    const int t = threadIdx.x;

    float sqsum = 0.f;
    for (int i = t; i < N_TOT; i += 256) sqsum += sq[i];
    red[t] = sqsum;
    __syncthreads();
    #pragma unroll
    for (int off = 128; off > 0; off >>= 1) {
        if (t < off) red[t] += red[t + off];
        __syncthreads();
    }
    const float tot_sq = red[0];
    __syncthreads();

    float m = 0.f;
    for (int b = 0; b < 64; ++b) m += partial[b * DIMS + t];
    red[t] = m * m;
    __syncthreads();
    #pragma unroll
    for (int off = 128; off > 0; off >>= 1) {
        if (t < off) red[t] += red[t + off];
        __syncthreads();
    }
    if (t == 0) {
        const double n = (double)N_TOT;
        double S  = 2.0 * n * (double)tot_sq - 2.0 * (double)red[0];
        double bw = S / (n * n - n) / 4.0;
        bw_out[0] = (float)bw;
    }
}

// ---------------------------------------------------------------------------
// Kernel 4: fused WMMA Gram + multi-kernel exp epilogue + signed reduce.
// Block = 256 thr = 8 waves arranged 4(row) x 2(col); block tile 64 x 128.
// Each wave computes a 16x64 strip = 4 x v_wmma_f32_16x16x4_f32 accumulators.
// Grid: (8192/128, 8192/64) = (64, 128).
// ---------------------------------------------------------------------------
__global__ void mmd_wmma_kernel(const float* __restrict__ s,
                                const float* __restrict__ t,
                                const float* __restrict__ sq,
                                const float* __restrict__ bwp,
                                float* __restrict__ blk_partial) {
    const int tid    = threadIdx.x;
    const int wave   = tid >> 5;
    const int lane   = tid & 31;
    const int half   = lane >> 4;     // 0: K=0,1   1: K=2,3 (A and B layout)
    const int lane16 = lane & 15;

    const int wr = wave & 3;          // 0..3
    const int wc = wave >> 2;         // 0..1
    const int rowBase = blockIdx.y * 64  + wr * 16;
    const int colBase = blockIdx.x * 128 + wc * 64;

    // Per-lane streaming pointers (float2 = the 2 K-values this lane owns).
    const float* aPtr = row_ptr(s, t, rowBase) + (size_t)lane16 * DIMS + 2 * half;
    const float* bPtr[4];
    #pragma unroll
    for (int j = 0; j < 4; ++j)
        bPtr[j] = row_ptr(s, t, colBase + j * 16) + (size_t)lane16 * DIMS + 2 * half;

    v8f acc[4];
    #pragma unroll
    for (int j = 0; j < 4; ++j) acc[j] = (v8f){0.f,0.f,0.f,0.f,0.f,0.f,0.f,0.f};

    #pragma unroll 4
    for (int k = 0; k < DIMS; k += 4) {
        v2f a = *(const v2f*)(aPtr + k);
        #pragma unroll
        for (int j = 0; j < 4; ++j) {
            v2f b = *(const v2f*)(bPtr[j] + k);
            acc[j] = __builtin_amdgcn_wmma_f32_16x16x4_f32(
                /*neg_a=*/false, a, /*neg_b=*/false, b,
                /*c_mod=*/(short)0, acc[j],
                /*reuse_a=*/false, /*reuse_b=*/false);
        }
    }

    // Epilogue: L2 = sq_i + sq_j - 2*dot; 5-scale Gaussian kernel; signed sum.
    const float bw = bwp[0];
    float rowSq[8];
    #pragma unroll
    for (int v = 0; v < 8; ++v)
        rowSq[v] = sq[rowBase + 8 * half + v];   // C layout: M = v + 8*half
    const bool sA = (rowBase < N_SRC);

    float local = 0.f;
    #pragma unroll
    for (int j = 0; j < 4; ++j) {
        const float colSq = sq[colBase + j * 16 + lane16];   // N = lane16
        const float sign  = (((colBase + j * 16) < N_SRC) == sA) ? 1.f : -1.f;
        float tilesum = 0.f;
        #pragma unroll
        for (int v = 0; v < 8; ++v) {
            const float l2 = rowSq[v] + colSq - 2.f * acc[j][v];
            float bwi = bw;
            float ks  = 0.f;
            #pragma unroll
            for (int i = 0; i < 5; ++i) {
                ks += __expf(-l2 / (bwi + EPS));
                bwi *= 2.f;                       // KERNEL_MUL
            }
            tilesum += ks;
        }
        local += sign * tilesum;
    }

    // wave reduce, then cross-wave via LDS, one partial per block.
    #pragma unroll
    for (int off = 16; off > 0; off >>= 1)
        local += __shfl_xor(local, off, 32);
    __shared__ float wred[8];
    if (lane == 0) wred[wave] = local;
    __syncthreads();
    if (tid == 0) {
        float bsum = 0.f;
        #pragma unroll
        for (int w = 0; w < 8; ++w) bsum += wred[w];
        blk_partial[blockIdx.y * gridDim.x + blockIdx.x] = bsum;
    }
}

// ---------------------------------------------------------------------------
// Kernel 5: sum 8192 block partials, divide by 4096^2, write scalar output.
// ---------------------------------------------------------------------------
__global__ void mmd_final_kernel(const float* __restrict__ blk_partial,
                                 int n_part, float* __restrict__ out) {
    __shared__ float red[256];
    const int t = threadIdx.x;
    float acc = 0.f;
    for (int i = t; i < n_part; i += 256) acc += blk_partial[i];
    red[t] = acc;
    __syncthreads();
    #pragma unroll
    for (int off = 128; off > 0; off >>= 1) {
        if (t < off) red[t] += red[t + off];
        __syncthreads();
    }
    if (t == 0) out[0] = red[0] / ((float)N_SRC * (float)N_SRC);
}

// ---------------------------------------------------------------------------
extern "C" void kernel_launch(void* const* d_in, const int* in_sizes, int n_in,
                              void* d_out, int out_size, void* d_ws, size_t ws_size,
                              hipStream_t stream) {
    const float* src = (const float*)d_in[0];   // 4096 x 256 f32
    const float* tgt = (const float*)d_in[1];   // 4096 x 256 f32
    float* out = (float*)d_out;

    float* ws       = (float*)d_ws;
    float* bw       = ws;                       // [1]
    float* col_part = ws + 64;                  // [64 * 256]
    float* sq       = col_part + 64 * DIMS;     // [8192]
    float* blk_part = sq + N_TOT;               // [64 * 128 = 8192]
    (void)ws_size; (void)in_sizes; (void)n_in; (void)out_size;

    mmd_colsum_kernel<<<64, 256, 0, stream>>>(src, tgt, col_part);
    mmd_rowsq_kernel<<<N_TOT / 8, 256, 0, stream>>>(src, tgt, sq);
    mmd_bw_kernel<<<1, 256, 0, stream>>>(col_part, sq, bw);
    dim3 grid(N_TOT / 128, N_TOT / 64);         // (64, 128)
    mmd_wmma_kernel<<<grid, 256, 0, stream>>>(src, tgt, sq, bw, blk_part);
    mmd_final_kernel<<<1, 256, 0, stream>>>(blk_part, 64 * 128, out);
}